// BDLModule_34488587387542
// MI455X (gfx1250) — compile-verified
//
#include <hip/hip_runtime.h>
#include <math.h>

typedef __attribute__((ext_vector_type(2))) float v2f;
typedef __attribute__((ext_vector_type(8))) float v8f;

#define DIM 128
#define HID 256
#define NB  8

// ---------------------------------------------------------------- utilities
__global__ void fill_zero_kernel(float* __restrict__ p, long count) {
    long i = (long)blockIdx.x * blockDim.x + threadIdx.x;
    if (i < count) p[i] = 0.0f;
}

// out-degree as float (exact for counts << 2^24)
__global__ void degree_kernel(const int* __restrict__ src,
                              float* __restrict__ deg, int nE) {
    int i = blockIdx.x * blockDim.x + threadIdx.x;
    if (i < nE) atomicAdd(deg + src[i], 1.0f);
}

// ---------------------------------------------------------------- forward rotation
// h0[n, b*16 + c*4 + e] = sum_d nr[n,b,c,d] * x[n, b*16 + d*4 + e]
__global__ void rotate_kernel(const float* __restrict__ x,
                              const float* __restrict__ node_rep,
                              float* __restrict__ h0, int n) {
    int t = blockIdx.x * blockDim.x + threadIdx.x;   // (node, bundle)
    if (t >= n * NB) return;
    int node = t >> 3, b = t & 7;
    const float4* R4 = (const float4*)(node_rep + (size_t)t * 16);
    const float4* V4 = (const float4*)(x + (size_t)node * DIM + b * 16);
    float R[16], V[16];
#pragma unroll
    for (int i = 0; i < 4; ++i) {
        float4 r = R4[i]; R[i*4+0]=r.x; R[i*4+1]=r.y; R[i*4+2]=r.z; R[i*4+3]=r.w;
        float4 v = V4[i]; V[i*4+0]=v.x; V[i*4+1]=v.y; V[i*4+2]=v.z; V[i*4+3]=v.w;
    }
    float* o = h0 + (size_t)node * DIM + b * 16;
#pragma unroll
    for (int c = 0; c < 4; ++c)
#pragma unroll
        for (int e = 0; e < 4; ++e)
            o[c*4+e] = R[c*4+0]*V[0*4+e] + R[c*4+1]*V[1*4+e]
                     + R[c*4+2]*V[2*4+e] + R[c*4+3]*V[3*4+e];
}

// ---------------------------------------------------------------- SpMM round
// one wave per edge; lane handles 4 consecutive features
__global__ void spmm_kernel(const float* __restrict__ hin,
                            const int* __restrict__ src,
                            const int* __restrict__ dst,
                            const float* __restrict__ deg,
                            float* __restrict__ hout, int nE) {
    int wid  = (int)(((long)blockIdx.x * blockDim.x + threadIdx.x) >> 5);
    int lane = threadIdx.x & 31;
    if (wid >= nE) return;
    int s = src[wid], d = dst[wid];
    float coef = rsqrtf(deg[s] * deg[d]);
    float4 v = *(const float4*)(hin + (size_t)s * DIM + lane * 4);
    float* o = hout + (size_t)d * DIM + lane * 4;
    atomicAdd(o + 0, v.x * coef);
    atomicAdd(o + 1, v.y * coef);
    atomicAdd(o + 2, v.z * coef);
    atomicAdd(o + 3, v.w * coef);
}

// ---------------------------------------------------------------- fused invrot + MLP
// 128 threads (4 waves) per 16-node tile.  n must be a multiple of 16 (100000 is).
__global__ __launch_bounds__(128)
void ffn_kernel(const float* __restrict__ h,
                const float* __restrict__ node_rep,
                const float* __restrict__ w1, const float* __restrict__ b1,
                const float* __restrict__ w2, const float* __restrict__ b2,
                float* __restrict__ out, int n) {
    __shared__ float sG[16 * 132];   // 16 rows x DIM, pad 4 (stride % 64 == 4)
    __shared__ float sH[16 * 260];   // 16 rows x HID, pad 4

    const int t  = threadIdx.x;
    const int n0 = blockIdx.x * 16;

    // stage: load 16 rows, applying transposed per-bundle rotation
    {
        const int r = t >> 3, b = t & 7;
        const int node = n0 + r;
        const float4* R4 = (const float4*)(node_rep + ((size_t)node * NB + b) * 16);
        const float4* V4 = (const float4*)(h + (size_t)node * DIM + b * 16);
        float R[16], V[16];
#pragma unroll
        for (int i = 0; i < 4; ++i) {
            float4 rv = R4[i]; R[i*4+0]=rv.x; R[i*4+1]=rv.y; R[i*4+2]=rv.z; R[i*4+3]=rv.w;
            float4 vv = V4[i]; V[i*4+0]=vv.x; V[i*4+1]=vv.y; V[i*4+2]=vv.z; V[i*4+3]=vv.w;
        }
        float* g = &sG[r * 132 + b * 16];
#pragma unroll
        for (int c = 0; c < 4; ++c)
#pragma unroll
            for (int e = 0; e < 4; ++e)   // einsum('nbdc,nbde->nbce'): R transposed
                g[c*4+e] = R[0*4+c]*V[0*4+e] + R[1*4+c]*V[1*4+e]
                         + R[2*4+c]*V[2*4+e] + R[3*4+c]*V[3*4+e];
    }
    __syncthreads();

    const int lane = t & 31;
    const int wave = t >> 5;
    const int m  = lane & 15;          // row-in-tile / col-in-tile per fragment rules
    const int kh = (lane >> 4) << 1;   // 0 or 2: K sub-offset for this half-wave

    // GEMM1: H = gelu(G @ W1^T + b1), 16x16 tiles over HID, K = DIM
    for (int nt = wave; nt < HID / 16; nt += 4) {
        const int col = nt * 16 + m;
        const float bias = b1[col];
        v8f acc;
#pragma unroll
        for (int j = 0; j < 8; ++j) acc[j] = bias;
        const float* w1r = w1 + (size_t)col * DIM;   // B[k][n] = W1[col][k]
#pragma unroll 4
        for (int k0 = 0; k0 < DIM; k0 += 4) {
            v2f a, bm;
            a.x  = sG[m * 132 + k0 + kh];
            a.y  = sG[m * 132 + k0 + kh + 1];
            bm.x = w1r[k0 + kh];
            bm.y = w1r[k0 + kh + 1];
            acc = __builtin_amdgcn_wmma_f32_16x16x4_f32(
                      false, a, false, bm, (short)0, acc, false, false);
        }
#pragma unroll
        for (int j = 0; j < 8; ++j) {  // C vgpr j -> row j + 8*(lane/16)
            float xv = acc[j];
            float gl = 0.5f * xv * (1.0f + erff(xv * 0.70710678118654752f));
            sH[(j + (kh << 2)) * 260 + col] = gl;
        }
    }
    __syncthreads();

    // GEMM2: OUT = H @ W2^T + b2, 16x16 tiles over DIM, K = HID
    for (int nt = wave; nt < DIM / 16; nt += 4) {
        const int col = nt * 16 + m;
        const float bias = b2[col];
        v8f acc;
#pragma unroll
        for (int j = 0; j < 8; ++j) acc[j] = bias;
        const float* w2r = w2 + (size_t)col * HID;
#pragma unroll 4
        for (int k0 = 0; k0 < HID; k0 += 4) {
            v2f a, bm;
            a.x  = sH[m * 260 + k0 + kh];
            a.y  = sH[m * 260 + k0 + kh + 1];
            bm.x = w2r[k0 + kh];
            bm.y = w2r[k0 + kh + 1];
            acc = __builtin_amdgcn_wmma_f32_16x16x4_f32(
                      false, a, false, bm, (short)0, acc, false, false);
        }
#pragma unroll
        for (int j = 0; j < 8; ++j)
            out[(size_t)(n0 + j + (kh << 2)) * DIM + col] = acc[j];
    }
}

// ---------------------------------------------------------------- launch
extern "C" void kernel_launch(void* const* d_in, const int* in_sizes, int n_in,
                              void* d_out, int out_size, void* d_ws, size_t ws_size,
                              hipStream_t stream) {
    const float* x  = (const float*)d_in[0];
    const float* nr = (const float*)d_in[1];
    const int*   src = (const int*)d_in[2];
    const int*   dst = (const int*)d_in[3];
    const float* w1 = (const float*)d_in[4];
    const float* b1 = (const float*)d_in[5];
    const float* w2 = (const float*)d_in[6];
    const float* b2 = (const float*)d_in[7];

    const int n  = in_sizes[0] / DIM;   // 100000
    const int nE = in_sizes[2];         // 1600000

    float* deg = (float*)d_ws;
    size_t degPad = ((size_t)n + 255) & ~(size_t)255;
    float* bufA = deg + degPad;         // [n, DIM] ping buffer
    float* hOut = (float*)d_out;        // doubles as the other ping buffer

    const long nfeat = (long)n * DIM;
    dim3 blk(256);

    // degrees
    fill_zero_kernel<<<dim3((n + 255) / 256), blk, 0, stream>>>(deg, n);
    degree_kernel<<<dim3((nE + 255) / 256), blk, 0, stream>>>(src, deg, nE);

    // forward rotation: x -> bufA (writes every element, no pre-zero needed)
    rotate_kernel<<<dim3((n * NB + 255) / 256), blk, 0, stream>>>(x, nr, bufA, n);

    // SpMM round 1: bufA -> d_out
    fill_zero_kernel<<<dim3((int)((nfeat + 255) / 256)), blk, 0, stream>>>(hOut, nfeat);
    spmm_kernel<<<dim3((int)(((long)nE * 32 + 255) / 256)), blk, 0, stream>>>(
        bufA, src, dst, deg, hOut, nE);

    // SpMM round 2: d_out -> bufA
    fill_zero_kernel<<<dim3((int)((nfeat + 255) / 256)), blk, 0, stream>>>(bufA, nfeat);
    spmm_kernel<<<dim3((int)(((long)nE * 32 + 255) / 256)), blk, 0, stream>>>(
        hOut, src, dst, deg, bufA, nE);

    // fused inverse rotation + MLP with f32 WMMA: bufA -> d_out
    ffn_kernel<<<dim3(n / 16), dim3(128), 0, stream>>>(
        bufA, nr, w1, b1, w2, b2, hOut, n);
}